// MemoryAccess_70463233458485
// MI455X (gfx1250) — compile-verified
//
#include <hip/hip_runtime.h>
#include <math.h>

// ---------------------------------------------------------------------------
// fp32 WMMA fragment types (CDNA5 gfx1250, wave32)
// A 16x4 f32  -> 2 VGPRs/lane -> v2f
// B 4x16 f32  -> 2 VGPRs/lane -> v2f
// C/D 16x16 f32 -> 8 VGPRs/lane -> v8f
// ---------------------------------------------------------------------------
typedef __attribute__((ext_vector_type(2))) float v2f;
typedef __attribute__((ext_vector_type(8))) float v8f;

#define NSLOTS 1000
#define RSLOTS 3
#define NB 8
#define ENC_DIM 128
#define RD_N (RSLOTS * NSLOTS)   // 3000
#define IMG_HW (64 * 64)          // 4096
#define CHW3 (3 * IMG_HW)         // 12288
#define CHW6 (6 * IMG_HW)         // 24576

// ---------------------------------------------------------------------------
// Generic direct conv (no bias), optional fused relu / elementwise-mul /
// per-batch scale (scale applied after relu).
// ---------------------------------------------------------------------------
__global__ void conv2d_kernel(const float* __restrict__ in,
                              const float* __restrict__ w,
                              float* __restrict__ out,
                              int B, int IC, int IH, int IW,
                              int OC, int OH, int OW,
                              int K, int stride, int pad,
                              int do_relu,
                              const float* __restrict__ mulsrc,
                              const float* __restrict__ bscale,
                              int bs_stride, int bs_off) {
    int i = blockIdx.x * blockDim.x + threadIdx.x;
    int total = B * OC * OH * OW;
    if (i >= total) return;
    int ow = i % OW; int t = i / OW;
    int oh = t % OH; t /= OH;
    int oc = t % OC; int b = t / OC;
    float acc = 0.0f;
    for (int ic = 0; ic < IC; ++ic) {
        const float* inp = in + ((size_t)(b * IC + ic)) * IH * IW;
        const float* wp  = w + ((size_t)(oc * IC + ic)) * K * K;
        for (int kh = 0; kh < K; ++kh) {
            int ih = oh * stride - pad + kh;
            if (ih < 0 || ih >= IH) continue;
            for (int kw = 0; kw < K; ++kw) {
                int iw = ow * stride - pad + kw;
                if (iw < 0 || iw >= IW) continue;
                acc = fmaf(inp[ih * IW + iw], wp[kh * K + kw], acc);
            }
        }
    }
    if (do_relu) acc = fmaxf(acc, 0.0f);
    if (mulsrc) acc *= mulsrc[i];
    if (bscale) acc *= bscale[b * bs_stride + bs_off];
    out[i] = acc;
}

// ---------------------------------------------------------------------------
// BatchNorm (stats over N,H,W per channel) + ReLU, in place. 1 block/channel.
// ---------------------------------------------------------------------------
__global__ void bn_relu_kernel(float* __restrict__ data,
                               const float* __restrict__ g,
                               const float* __restrict__ bt,
                               int B, int C, int HW) {
    int c = blockIdx.x;
    int t = threadIdx.x;           // 256 threads
    int n = B * HW;
    float s = 0.0f, s2 = 0.0f;
    for (int i = t; i < n; i += 256) {
        int b = i / HW, r = i % HW;
        float v = data[((size_t)(b * C + c)) * HW + r];
        s += v; s2 += v * v;
    }
    __shared__ float sh[256], sh2[256];
    sh[t] = s; sh2[t] = s2; __syncthreads();
    for (int o = 128; o > 0; o >>= 1) {
        if (t < o) { sh[t] += sh[t + o]; sh2[t] += sh2[t + o]; }
        __syncthreads();
    }
    float mean = sh[0] / n;
    float var  = sh2[0] / n - mean * mean;
    float inv  = rsqrtf(var + 1e-5f);
    float gg = g[c], bb = bt[c];
    for (int i = t; i < n; i += 256) {
        int b = i / HW, r = i % HW;
        size_t off = ((size_t)(b * C + c)) * HW + r;
        float v = (data[off] - mean) * inv * gg + bb;
        data[off] = fmaxf(v, 0.0f);
    }
}

// ---------------------------------------------------------------------------
// fast_attn on enc (8x128): a = softmax(enc@w1^T + b1), out = enc*(a@w2^T+b2)
// 1 block per batch row, 128 threads.
// ---------------------------------------------------------------------------
__global__ void fast_attn_kernel(const float* __restrict__ enc,
                                 const float* __restrict__ w1,
                                 const float* __restrict__ b1,
                                 const float* __restrict__ w2,
                                 const float* __restrict__ b2,
                                 float* __restrict__ out) {
    int b = blockIdx.x;
    int t = threadIdx.x;           // 0..127
    __shared__ float xs[ENC_DIM], as[ENC_DIM], red[ENC_DIM];
    xs[t] = enc[b * ENC_DIM + t];
    __syncthreads();
    float v = b1[t];
    for (int k = 0; k < ENC_DIM; ++k) v = fmaf(xs[k], w1[t * ENC_DIM + k], v);
    // softmax over the 128 features
    red[t] = v; __syncthreads();
    for (int o = 64; o > 0; o >>= 1) {
        if (t < o) red[t] = fmaxf(red[t], red[t + o]);
        __syncthreads();
    }
    float mx = red[0]; __syncthreads();
    float e = expf(v - mx);
    red[t] = e; __syncthreads();
    for (int o = 64; o > 0; o >>= 1) {
        if (t < o) red[t] += red[t + o];
        __syncthreads();
    }
    float a = e / red[0];
    as[t] = a; __syncthreads();
    float u = b2[t];
    for (int k = 0; k < ENC_DIM; ++k) u = fmaf(as[k], w2[t * ENC_DIM + k], u);
    out[b * ENC_DIM + t] = xs[t] * u;
}

// ---------------------------------------------------------------------------
// read = tanh(fa @ rd_w^T + rd_b)  -- M=8 (padded to 16), K=128, N=3000.
// One wave computes FOUR 16-wide N tiles via V_WMMA_F32_16X16X4_F32.
//
// K is processed in chunks of 16 (4 WMMA K-steps).  Per chunk we first issue
// ALL 20 fragment loads (4 A + 16 B) into live register arrays, then the 16
// WMMAs.  Because every loaded value is consumed only after the full load
// block, the register allocator must keep them simultaneously live, letting
// loads retire under preceding WMMAs (partial loadcnt waits instead of the
// load->wait(0)->wmma serialization).  __launch_bounds__(32,1) grants the
// full VGPR budget so the scheduler has no pressure incentive to serialize.
//
// All loads are UNCONDITIONAL: out-of-range tile rows (m=8..15) and tail
// columns (n>=3000) are never stored, so their inputs are don't-care and we
// simply clamp the load indices (no EXEC divergence around WMMA, per the ISA
// requirement that WMMA executes with EXEC all-ones).
//
// A lane layout: M=lane&15, K=k0+2*(lane>>4)+{0,1}; B symmetric (N=lane&15).
// D layout: VGPR v -> M=v+8*(lane>>4), N=lane&15.
// ---------------------------------------------------------------------------
#define GEMM_TILES 4
#define KSTEPS_PER_CHUNK 4          // 4 x (K=4) = 16 K per chunk
__global__ __launch_bounds__(32, 1) void read_gemm_wmma_kernel(
        const float* __restrict__ fa,    // (8,128)
        const float* __restrict__ rdw,   // (3000,128)
        const float* __restrict__ rdb,   // (3000)
        float* __restrict__ readout) {   // (8,3000)
    const int lane = threadIdx.x;
    const int lm   = lane & 15;
    const int half = lane >> 4;
    const int nb0  = blockIdx.x * (16 * GEMM_TILES);

    const v2f* __restrict__ fa2 = (const v2f*)fa;    // row stride 64 v2f
    const v2f* __restrict__ rw2 = (const v2f*)rdw;   // row stride 64 v2f

    const int arow = lm & 7;                          // clamp rows 8..15 -> 0..7
    size_t brow[GEMM_TILES];
    #pragma unroll
    for (int t = 0; t < GEMM_TILES; ++t) {
        int n = nb0 + t * 16 + lm;
        brow[t] = (size_t)((n < RD_N) ? n : (RD_N - 1)) * 64;  // clamp tail
    }

    // warm the cache for this wave's 4 B rows (512 B per row per lane)
    #pragma unroll
    for (int t = 0; t < GEMM_TILES; ++t)
        __builtin_prefetch((const void*)(rw2 + brow[t]), 0, 1);

    v8f acc[GEMM_TILES];
    #pragma unroll
    for (int t = 0; t < GEMM_TILES; ++t) acc[t] = (v8f){};

    #pragma unroll 1
    for (int kc = 0; kc < ENC_DIM; kc += 4 * KSTEPS_PER_CHUNK) {
        // ---- bulk load phase: 20 independent b64 loads, all live ----
        v2f afr[KSTEPS_PER_CHUNK];
        v2f bfr[KSTEPS_PER_CHUNK][GEMM_TILES];
        #pragma unroll
        for (int s = 0; s < KSTEPS_PER_CHUNK; ++s) {
            const int kv = ((kc + 4 * s) >> 1) + half;  // v2f index in a row
            afr[s] = fa2[arow * 64 + kv];
            #pragma unroll
            for (int t = 0; t < GEMM_TILES; ++t)
                bfr[s][t] = rw2[brow[t] + kv];
        }
        // ---- matrix phase: 16 WMMAs ----
        #pragma unroll
        for (int s = 0; s < KSTEPS_PER_CHUNK; ++s) {
            #pragma unroll
            for (int t = 0; t < GEMM_TILES; ++t)
                acc[t] = __builtin_amdgcn_wmma_f32_16x16x4_f32(
                    false, afr[s], false, bfr[s][t],
                    (short)0, acc[t], false, false);
        }
    }

    #pragma unroll
    for (int t = 0; t < GEMM_TILES; ++t) {
        int n = nb0 + t * 16 + lm;
        if (n < RD_N && half == 0) {                  // rows 0..7 live in half 0
            float bias = rdb[n];
            #pragma unroll
            for (int v = 0; v < 8; ++v)
                readout[(size_t)v * RD_N + n] = tanhf(acc[t][v] + bias);
        }
    }
}

// ---------------------------------------------------------------------------
// Per (b,j): max + argmax over the 1000 slots. 1 block per (b,j), 128 thr.
// ---------------------------------------------------------------------------
__global__ void maxidx_kernel(const float* __restrict__ readv,
                              float* __restrict__ rw,
                              int* __restrict__ idx) {
    int bj = blockIdx.x;                 // 0..23 : b = bj/3, j = bj%3
    const float* p = readv + (size_t)(bj / RSLOTS) * RD_N + (bj % RSLOTS) * NSLOTS;
    int t = threadIdx.x;                 // 128
    float best = -3.402823466e+38f; int bi = 0;
    for (int s = t; s < NSLOTS; s += 128) {
        float v = p[s];
        if (v > best) { best = v; bi = s; }
    }
    __shared__ float sv[128]; __shared__ int si[128];
    sv[t] = best; si[t] = bi; __syncthreads();
    for (int o = 64; o > 0; o >>= 1) {
        if (t < o) {
            if (sv[t + o] > sv[t] || (sv[t + o] == sv[t] && si[t + o] < si[t])) {
                sv[t] = sv[t + o]; si[t] = si[t + o];
            }
        }
        __syncthreads();
    }
    if (t == 0) { rw[bj] = sv[0]; idx[bj] = si[0]; }
}

// uw[b,j] = sigmoid(fa_u[b,:] . uw_w[j,:] + uw_b[j])
__global__ void uw_kernel(const float* __restrict__ fa_u,
                          const float* __restrict__ uw_w,
                          const float* __restrict__ uw_b,
                          float* __restrict__ uw) {
    int t = threadIdx.x;
    if (t >= NB * RSLOTS) return;
    int b = t / RSLOTS, j = t % RSLOTS;
    float s = uw_b[j];
    for (int k = 0; k < ENC_DIM; ++k)
        s = fmaf(fa_u[b * ENC_DIM + k], uw_w[j * ENC_DIM + k], s);
    uw[t] = 1.0f / (1.0f + expf(-s));
}

// r[b,c,h,w] = memory[b, idx[b,j], c, h, w]
__global__ void gather_kernel(const float* __restrict__ mem,
                              const int* __restrict__ idx, int j,
                              float* __restrict__ r) {
    int i = blockIdx.x * blockDim.x + threadIdx.x;
    if (i >= NB * CHW3) return;
    int b = i / CHW3, rest = i % CHW3;
    size_t slot = (size_t)b * NSLOTS + idx[b * RSLOTS + j];
    r[i] = mem[slot * CHW3 + rest];
}

// dst(B,2C,HW) = concat(s0(B,C,HW), s1(B,C,HW)) along channel dim
__global__ void concat2_kernel(const float* __restrict__ s0,
                               const float* __restrict__ s1,
                               float* __restrict__ dst,
                               int B, int C, int HW) {
    int i = blockIdx.x * blockDim.x + threadIdx.x;
    int total = B * 2 * C * HW;
    if (i >= total) return;
    int hw = i % HW; int t = i / HW;
    int c = t % (2 * C); int b = t / (2 * C);
    dst[i] = (c < C) ? s0[((size_t)b * C + c) * HW + hw]
                     : s1[((size_t)b * C + (c - C)) * HW + hw];
}

// in-place softmax over the H axis of (B,C,H,W); one thread per (b,c,w) column
__global__ void softmax_h_kernel(float* __restrict__ a, int B, int C, int H, int W) {
    int col = blockIdx.x * blockDim.x + threadIdx.x;
    if (col >= B * C * W) return;
    int w = col % W; int t = col / W;
    int c = t % C; int b = t / C;
    float* p = a + (((size_t)b * C + c) * H) * W + w;
    float mx = -3.402823466e+38f;
    for (int h = 0; h < H; ++h) mx = fmaxf(mx, p[(size_t)h * W]);
    float s = 0.0f;
    for (int h = 0; h < H; ++h) s += expf(p[(size_t)h * W] - mx);
    float inv = 1.0f / s;
    for (int h = 0; h < H; ++h) p[(size_t)h * W] = expf(p[(size_t)h * W] - mx) * inv;
}

// r = w*u + (1-w)*r  with w = uw[b*3+j]; in place on r
__global__ void lerp_kernel(const float* __restrict__ u,
                            float* __restrict__ r,
                            const float* __restrict__ uw, int j) {
    int i = blockIdx.x * blockDim.x + threadIdx.x;
    if (i >= NB * CHW3) return;
    int b = i / CHW3;
    float w = uw[b * RSLOTS + j];
    r[i] = w * u[i] + (1.0f - w) * r[i];
}

__global__ void zero_kernel(float* __restrict__ p, int n) {
    int i = blockIdx.x * blockDim.x + threadIdx.x;
    if (i < n) p[i] = 0.0f;
}

__global__ void tanh_kernel(const float* __restrict__ in, float* __restrict__ out, int n) {
    int i = blockIdx.x * blockDim.x + threadIdx.x;
    if (i < n) out[i] = tanhf(in[i]);
}

// ---------------------------------------------------------------------------
// host-side launcher
// ---------------------------------------------------------------------------
static inline void launch_conv(const float* in, const float* w, float* out,
                               int B, int IC, int IH, int IW,
                               int OC, int OH, int OW, int K, int s, int p,
                               int relu, const float* mulsrc,
                               const float* bscale, int bss, int bso,
                               hipStream_t st) {
    int total = B * OC * OH * OW;
    conv2d_kernel<<<(total + 255) / 256, 256, 0, st>>>(
        in, w, out, B, IC, IH, IW, OC, OH, OW, K, s, p,
        relu, mulsrc, bscale, bss, bso);
}

extern "C" void kernel_launch(void* const* d_in, const int* in_sizes, int n_in,
                              void* d_out, int out_size, void* d_ws, size_t ws_size,
                              hipStream_t stream) {
    const float* inputs = (const float*)d_in[0];
    const float* memory = (const float*)d_in[1];
    const float* enc_w1 = (const float*)d_in[2];
    const float* enc_g1 = (const float*)d_in[3];
    const float* enc_b1 = (const float*)d_in[4];
    const float* enc_w2 = (const float*)d_in[5];
    const float* enc_g2 = (const float*)d_in[6];
    const float* enc_b2 = (const float*)d_in[7];
    const float* enc_w3 = (const float*)d_in[8];
    const float* enc_g3 = (const float*)d_in[9];
    const float* enc_b3 = (const float*)d_in[10];
    const float* enc_w4 = (const float*)d_in[11];
    const float* enc_g4 = (const float*)d_in[12];
    const float* enc_b4 = (const float*)d_in[13];
    const float* rfa_w1 = (const float*)d_in[14];
    const float* rfa_b1 = (const float*)d_in[15];
    const float* rfa_w2 = (const float*)d_in[16];
    const float* rfa_b2 = (const float*)d_in[17];
    const float* rd_w   = (const float*)d_in[18];
    const float* rd_b   = (const float*)d_in[19];
    const float* ufa_w1 = (const float*)d_in[20];
    const float* ufa_b1 = (const float*)d_in[21];
    const float* ufa_w2 = (const float*)d_in[22];
    const float* ufa_b2 = (const float*)d_in[23];
    const float* uw_w   = (const float*)d_in[24];
    const float* uw_b   = (const float*)d_in[25];
    const float* um_fa1 = (const float*)d_in[26];
    const float* um_fa2 = (const float*)d_in[27];
    const float* um_w   = (const float*)d_in[28];
    const float* am_fa1 = (const float*)d_in[29];
    const float* am_fa2 = (const float*)d_in[30];
    const float* am_w   = (const float*)d_in[31];
    float* out = (float*)d_out;

    // workspace layout (floats)
    float* ws   = (float*)d_ws;
    float* h1   = ws;                 // 8*32*16*16 = 65536
    float* h2   = h1 + 65536;         // 8*64*8*8   = 32768
    float* h3   = h2 + 32768;         // 8*64*4*4   = 8192
    float* enc  = h3 + 8192;          // 8*128
    float* fa_r = enc + 1024;         // 8*128
    float* fa_u = fa_r + 1024;        // 8*128
    float* rdv  = fa_u + 1024;        // 8*3000 = 24000
    float* rw   = rdv + 24000;        // 24 (+pad)
    float* uw   = rw + 32;            // 24 (+pad)
    int*   idx  = (int*)(uw + 32);    // 24 (+pad)
    float* rbuf = uw + 64;            // 8*3*4096 = 98304
    float* xbuf = rbuf + NB * CHW3;   // 8*6*4096 = 196608
    float* abuf = xbuf + NB * CHW6;   // 196608
    float* ybuf = abuf + NB * CHW6;   // 196608
    float* ubuf = ybuf + NB * CHW6;   // 98304
    float* mbuf = ubuf + NB * CHW3;   // 98304
    (void)ws_size; (void)n_in; (void)in_sizes; (void)out_size;

    // ---- encoder ----
    launch_conv(inputs, enc_w1, h1, 8, 3, 64, 64, 32, 16, 16, 6, 4, 1, 0, nullptr, nullptr, 0, 0, stream);
    bn_relu_kernel<<<32, 256, 0, stream>>>(h1, enc_g1, enc_b1, 8, 32, 256);
    launch_conv(h1, enc_w2, h2, 8, 32, 16, 16, 64, 8, 8, 4, 2, 1, 0, nullptr, nullptr, 0, 0, stream);
    bn_relu_kernel<<<64, 256, 0, stream>>>(h2, enc_g2, enc_b2, 8, 64, 64);
    launch_conv(h2, enc_w3, h3, 8, 64, 8, 8, 64, 4, 4, 4, 2, 1, 0, nullptr, nullptr, 0, 0, stream);
    bn_relu_kernel<<<64, 256, 0, stream>>>(h3, enc_g3, enc_b3, 8, 64, 16);
    launch_conv(h3, enc_w4, enc, 8, 64, 4, 4, 128, 1, 1, 4, 1, 0, 0, nullptr, nullptr, 0, 0, stream);
    bn_relu_kernel<<<128, 256, 0, stream>>>(enc, enc_g4, enc_b4, 8, 128, 1);

    // ---- read head: fast_attn + WMMA GEMM (8x128 @ 128x3000) + tanh ----
    fast_attn_kernel<<<NB, ENC_DIM, 0, stream>>>(enc, rfa_w1, rfa_b1, rfa_w2, rfa_b2, fa_r);
    read_gemm_wmma_kernel<<<(RD_N + 16 * GEMM_TILES - 1) / (16 * GEMM_TILES), 32, 0, stream>>>(
        fa_r, rd_w, rd_b, rdv);
    maxidx_kernel<<<NB * RSLOTS, 128, 0, stream>>>(rdv, rw, idx);

    // ---- update gate ----
    fast_attn_kernel<<<NB, ENC_DIM, 0, stream>>>(enc, ufa_w1, ufa_b1, ufa_w2, ufa_b2, fa_u);
    uw_kernel<<<1, 32, 0, stream>>>(fa_u, uw_w, uw_b, uw);

    // ---- memory write loop ----
    int n3 = NB * CHW3, n6 = NB * CHW6;
    zero_kernel<<<(n3 + 255) / 256, 256, 0, stream>>>(mbuf, n3);
    for (int j = 0; j < RSLOTS; ++j) {
        gather_kernel<<<(n3 + 255) / 256, 256, 0, stream>>>(memory, idx, j, rbuf);
        // u = relu(conv(fast_attn_img([r, inputs]), um_w))
        concat2_kernel<<<(n6 + 255) / 256, 256, 0, stream>>>(rbuf, inputs, xbuf, 8, 3, IMG_HW);
        launch_conv(xbuf, um_fa1, abuf, 8, 6, 64, 64, 6, 64, 64, 3, 1, 1, 0, nullptr, nullptr, 0, 0, stream);
        softmax_h_kernel<<<(8 * 6 * 64 + 255) / 256, 256, 0, stream>>>(abuf, 8, 6, 64, 64);
        launch_conv(abuf, um_fa2, ybuf, 8, 6, 64, 64, 6, 64, 64, 3, 1, 1, 0, xbuf, nullptr, 0, 0, stream);
        launch_conv(ybuf, um_w, ubuf, 8, 6, 64, 64, 3, 64, 64, 3, 1, 1, 1, nullptr, nullptr, 0, 0, stream);
        // r = w*u + (1-w)*r
        lerp_kernel<<<(n3 + 255) / 256, 256, 0, stream>>>(ubuf, rbuf, uw, j);
        // m = rw * relu(conv(fast_attn_img([r, m]), am_w))
        concat2_kernel<<<(n6 + 255) / 256, 256, 0, stream>>>(rbuf, mbuf, xbuf, 8, 3, IMG_HW);
        launch_conv(xbuf, am_fa1, abuf, 8, 6, 64, 64, 6, 64, 64, 3, 1, 1, 0, nullptr, nullptr, 0, 0, stream);
        softmax_h_kernel<<<(8 * 6 * 64 + 255) / 256, 256, 0, stream>>>(abuf, 8, 6, 64, 64);
        launch_conv(abuf, am_fa2, ybuf, 8, 6, 64, 64, 6, 64, 64, 3, 1, 1, 0, xbuf, nullptr, 0, 0, stream);
        launch_conv(ybuf, am_w, mbuf, 8, 6, 64, 64, 3, 64, 64, 3, 1, 1, 1, nullptr, rw, RSLOTS, j, stream);
    }
    tanh_kernel<<<(n3 + 255) / 256, 256, 0, stream>>>(mbuf, out, n3);
}